// Renderer_65111704207925
// MI455X (gfx1250) — compile-verified
//
#include <hip/hip_runtime.h>

#define SH_C0 0.28209479177387814f
#define LOWPASS 0.3f
#define NEAR_CULL 0.2f
#define PLD 12  // floats per gaussian payload row: F0..F5, colR,colG,colB, opac, tz, pad

typedef __attribute__((ext_vector_type(2))) float v2f;
typedef __attribute__((ext_vector_type(4))) float v4f;
typedef __attribute__((ext_vector_type(8))) float v8f;

#if __has_builtin(__builtin_amdgcn_tensor_load_to_lds) && \
    __has_builtin(__builtin_amdgcn_s_wait_tensorcnt)
#define HAS_TDM 1
#else
#define HAS_TDM 0
#endif

// ---------------------------------------------------------------------------
// Phase 1: per-gaussian projection -> rank-6 power coefficients + payload
// ---------------------------------------------------------------------------
__global__ void preprocess_kernel(const float* __restrict__ extr,
                                  const float* __restrict__ intr,
                                  const float* __restrict__ nearv,
                                  const int* __restrict__ hPtr,
                                  const int* __restrict__ wPtr,
                                  const float* __restrict__ means,
                                  const float* __restrict__ covs,
                                  const float* __restrict__ shs,
                                  const float* __restrict__ opac,
                                  float* __restrict__ payload,
                                  float* __restrict__ keys,
                                  int B, int G) {
  int tid = blockIdx.x * blockDim.x + threadIdx.x;
  if (tid >= B * G) return;
  int b = tid / G, g = tid % G;
  float h = (float)hPtr[0], w = (float)wPtr[0];
  float scale = 1.0f / nearv[b];

  // extrinsics row-major 4x4; scale translation column
  const float* E = extr + b * 16;
  float A00 = E[0], A01 = E[1], A02 = E[2],  t0 = E[3] * scale;
  float A10 = E[4], A11 = E[5], A12 = E[6],  t1 = E[7] * scale;
  float A20 = E[8], A21 = E[9], A22 = E[10], t2 = E[11] * scale;

  // view = inv(affine extr): R = inv3x3(A), t = -R * t
  float m00 = A11 * A22 - A12 * A21;
  float m10 = A12 * A20 - A10 * A22;
  float m20 = A10 * A21 - A11 * A20;
  float det3 = A00 * m00 + A01 * m10 + A02 * m20;
  float id = 1.0f / det3;
  float R00 = m00 * id;
  float R01 = (A02 * A21 - A01 * A22) * id;
  float R02 = (A01 * A12 - A02 * A11) * id;
  float R10 = m10 * id;
  float R11 = (A00 * A22 - A02 * A20) * id;
  float R12 = (A02 * A10 - A00 * A12) * id;
  float R20 = m20 * id;
  float R21 = (A01 * A20 - A00 * A21) * id;
  float R22 = (A00 * A11 - A01 * A10) * id;
  float tx = -(R00 * t0 + R01 * t1 + R02 * t2);
  float ty = -(R10 * t0 + R11 * t1 + R12 * t2);
  float tzc = -(R20 * t0 + R21 * t1 + R22 * t2);

  const float* I = intr + b * 9;
  float fx = I[0], fy = I[4], cx = I[2], cy = I[5];
  float fx_p = fx * w, fy_p = fy * h, cx_p = cx * w, cy_p = cy * h;
  float lim_x = 1.3f * (0.5f / fx), lim_y = 1.3f * (0.5f / fy);

  const float* mp = means + (size_t)(b * G + g) * 3;
  float mX = mp[0] * scale, mY = mp[1] * scale, mZ = mp[2] * scale;
  float camx = R00 * mX + R01 * mY + R02 * mZ + tx;
  float camy = R10 * mX + R11 * mY + R12 * mZ + ty;
  float camz = R20 * mX + R21 * mY + R22 * mZ + tzc;
  float tz = camz;
  float inv_tz = 1.0f / tz;
  float txz = fminf(fmaxf(camx * inv_tz, -lim_x), lim_x) * tz;
  float tyz = fminf(fmaxf(camy * inv_tz, -lim_y), lim_y) * tz;

  float J00 = fx_p * inv_tz, J02 = -fx_p * txz * inv_tz * inv_tz;
  float J11 = fy_p * inv_tz, J12 = -fy_p * tyz * inv_tz * inv_tz;

  // M = J * R (2x3)
  float M00 = J00 * R00 + J02 * R20;
  float M01 = J00 * R01 + J02 * R21;
  float M02 = J00 * R02 + J02 * R22;
  float M10 = J11 * R10 + J12 * R20;
  float M11 = J11 * R11 + J12 * R21;
  float M12 = J11 * R12 + J12 * R22;

  const float* C = covs + (size_t)(b * G + g) * 9;
  float s2 = scale * scale;
  float S00 = C[0] * s2, S01 = C[1] * s2, S02 = C[2] * s2;
  float S10 = C[3] * s2, S11 = C[4] * s2, S12 = C[5] * s2;
  float S20 = C[6] * s2, S21 = C[7] * s2, S22 = C[8] * s2;

  // cov2d = M * S * M^T
  float T00 = M00 * S00 + M01 * S10 + M02 * S20;
  float T01 = M00 * S01 + M01 * S11 + M02 * S21;
  float T02 = M00 * S02 + M01 * S12 + M02 * S22;
  float T10 = M10 * S00 + M11 * S10 + M12 * S20;
  float T11 = M10 * S01 + M11 * S11 + M12 * S21;
  float T12 = M10 * S02 + M11 * S12 + M12 * S22;
  float V00 = T00 * M00 + T01 * M01 + T02 * M02;
  float V01 = T00 * M10 + T01 * M11 + T02 * M12;
  float V11 = T10 * M10 + T11 * M11 + T12 * M12;

  float a = V00 + LOWPASS, bb = V01, c2 = V11 + LOWPASS;
  float det = a * c2 - bb * bb;
  float inv_det = (det != 0.0f) ? 1.0f / det : 0.0f;
  float conA = c2 * inv_det, conB = -bb * inv_det, conC = a * inv_det;

  float mx = fx_p * camx * inv_tz + cx_p - 0.5f;
  float my = fy_p * camy * inv_tz + cy_p - 0.5f;

  const float* sh = shs + (size_t)(b * G + g) * 3;
  float colR = fmaxf(SH_C0 * sh[0] + 0.5f, 0.0f);
  float colG = fmaxf(SH_C0 * sh[1] + 0.5f, 0.0f);
  float colB = fmaxf(SH_C0 * sh[2] + 0.5f, 0.0f);

  bool valid = (tz > NEAR_CULL) && (det > 0.0f);
  float opv = valid ? opac[b * G + g] : 0.0f;

  // rank-6 decomposition of -0.5*(A dx^2 + C dy^2) - B dx dy
  float* P = payload + (size_t)(b * G + g) * PLD;
  P[0] = -0.5f * conA;                                               // * px^2
  P[1] = -0.5f * conC;                                               // * py^2
  P[2] = -conB;                                                      // * px*py
  P[3] = conA * mx + conB * my;                                      // * px
  P[4] = conC * my + conB * mx;                                      // * py
  P[5] = -0.5f * (conA * mx * mx + conC * my * my) - conB * mx * my; // * 1
  P[6] = colR; P[7] = colG; P[8] = colB;
  P[9] = opv;  P[10] = tz;  P[11] = 0.0f;

  keys[b * G + g] = valid ? tz : __builtin_inff();
}

// ---------------------------------------------------------------------------
// Phase 2: per-batch bitonic argsort by depth + gather into sorted payload
// ---------------------------------------------------------------------------
__global__ void sort_kernel(const float* __restrict__ keys,
                            const float* __restrict__ payload,
                            float* __restrict__ sorted, int G) {
  __shared__ float sk[1024];
  __shared__ int si[1024];
  int b = blockIdx.x;
  int t = threadIdx.x;
  int N = blockDim.x;  // pow2, <= 1024
  sk[t] = (t < G) ? keys[b * G + t] : __builtin_inff();
  si[t] = t;
  __syncthreads();
  for (int k = 2; k <= N; k <<= 1) {
    for (int j = k >> 1; j > 0; j >>= 1) {
      int ixj = t ^ j;
      if (ixj > t) {
        bool up = ((t & k) == 0);
        float av = sk[t], bv = sk[ixj];
        bool swap = up ? (av > bv) : (av < bv);
        if (swap) {
          sk[t] = bv; sk[ixj] = av;
          int tmp = si[t]; si[t] = si[ixj]; si[ixj] = tmp;
        }
      }
      __syncthreads();
    }
  }
  int src = si[t];
  float* dst = sorted + ((size_t)b * N + t) * PLD;
  if (src < G) {
    const float* s = payload + ((size_t)b * G + src) * PLD;
#pragma unroll
    for (int i = 0; i < PLD; ++i) dst[i] = s[i];
  } else {
#pragma unroll
    for (int i = 0; i < PLD; ++i) dst[i] = 0.0f;
  }
}

// ---------------------------------------------------------------------------
// TDM: DMA one 16-gaussian chunk (192 dwords, 768 B) global -> LDS.
// D# bitfields per CDNA5 ISA.
// 6-arg builtin: (uint32x4 g0, int32x8 g1, int32x4 g2, int32x4 g3,
//                 int32x8 extra, i32 cpol)
// ---------------------------------------------------------------------------
#if HAS_TDM
typedef __attribute__((ext_vector_type(4))) unsigned int v4u;
typedef __attribute__((ext_vector_type(8))) int v8i;
typedef __attribute__((ext_vector_type(4))) int v4i;

__device__ __forceinline__ void tdm_load_chunk(const float* gsrc, float* ldsDst) {
  unsigned long long ga = (unsigned long long)(size_t)gsrc;
  unsigned ga_lo = (unsigned)__builtin_amdgcn_readfirstlane((int)(unsigned)ga);
  unsigned ga_hi = (unsigned)__builtin_amdgcn_readfirstlane((int)(unsigned)(ga >> 32));
  unsigned laddr = (unsigned)__builtin_amdgcn_readfirstlane((int)(unsigned)(size_t)ldsDst);

  v4u g0;
  g0[0] = 1u;                                        // count=1, user mode
  g0[1] = laddr;                                     // lds_addr (bytes)
  g0[2] = ga_lo;                                     // global_addr[31:0]
  g0[3] = (ga_hi & 0x01FFFFFFu) | 0x80000000u;       // global_addr[56:32] | type=2

  v8i g1;
  g1[0] = 0x00020000;        // workgroup_mask=0 (no cluster), data_size=2 -> 4B
  g1[1] = (int)(192u << 16); // tensor_dim0[15:0] = 192 dwords
  g1[2] = (int)(1u << 16);   // tensor_dim0[31:16]=0 | tensor_dim1[15:0]=1
  g1[3] = (int)(192u << 16); // tensor_dim1[31:16]=0 | tile_dim0 = 192
  g1[4] = 0;                 // tile_dim1=0 (unused), tile_dim2=0 -> 1-D tile
  g1[5] = 192;               // tensor_dim0_stride[31:0]
  g1[6] = 0;                 // stride0[47:32]=0 | stride1[15:0]=0
  g1[7] = 0;                 // stride1[47:16]=0

  v4i z4 = {0, 0, 0, 0};               // groups 2/3: dims 2..4 unused
  v8i z8 = {0, 0, 0, 0, 0, 0, 0, 0};   // extra descriptor slot (unused)
  __builtin_amdgcn_tensor_load_to_lds(g0, g1, z4, z4, z8, 0);
}
#endif

// ---------------------------------------------------------------------------
// Phase 3: WMMA-tiled compositor. Wave = 16 pixels; 16-gaussian chunks.
// power tile D(16x16) = F(16x6) x H(6x16) via two v_wmma_f32_16x16x4_f32.
// Chunk payload double-buffered in LDS, filled by the Tensor Data Mover.
// ---------------------------------------------------------------------------
__global__ void render_kernel(const float* __restrict__ sorted,
                              const float* __restrict__ bg,
                              const int* __restrict__ wPtr,
                              float* __restrict__ out_img,
                              float* __restrict__ out_depth,
                              float* __restrict__ out_op,
                              int N, int P, int blocksPerBatch) {
  __shared__ float buf[2][16 * PLD];  // double-buffered 768B chunk slices
  int wave = threadIdx.x >> 5;
  int lane = threadIdx.x & 31;
  int b = blockIdx.x / blocksPerBatch;
  int tileBase = (blockIdx.x % blocksPerBatch) * 128 + wave * 16;
  int n = lane & 15;
  bool hi = lane >= 16;
  int p = tileBase + n;
  bool active = p < P;
  int pc = active ? p : (P - 1);
  int W = wPtr[0];
  float px = (float)(pc % W);
  float py = (float)(pc / W);

  // B fragments: H rows K0..K5 = [px^2, py^2, px*py, px, py, 1], K6,K7 = 0
  v2f b1, b2;
  b1.x = hi ? px * py : px * px;  // K2 | K0
  b1.y = hi ? px      : py * py;  // K3 | K1
  b2.x = hi ? 0.0f    : py;       // K6 | K4
  b2.y = hi ? 0.0f    : 1.0f;     // K7 | K5

  const float* base = sorted + (size_t)b * N * PLD;

  float T = 1.0f;
  float accR = 0, accG = 0, accB = 0, accD = 0, accW = 0;
  int chunks = N >> 4;
  int rbase = hi ? 8 : 0;

#if HAS_TDM
  if (wave == 0) {
    tdm_load_chunk(base, &buf[0][0]);
    __builtin_amdgcn_s_wait_tensorcnt(0);
  }
  __syncthreads();
#endif

  for (int c = 0; c < chunks; ++c) {
    int cb = c & 1;
#if HAS_TDM
    // TDM prefetch of next chunk overlaps this chunk's compute
    if (wave == 0 && (c + 1) < chunks)
      tdm_load_chunk(base + (size_t)(c + 1) * 16 * PLD, &buf[cb ^ 1][0]);
    const float* myLds = &buf[cb][0];
#else
    // fallback: cooperative staging (single buffer)
    {
      const float* gsrc = base + (size_t)c * 16 * PLD;
      if (threadIdx.x < 16 * PLD) buf[0][threadIdx.x] = gsrc[threadIdx.x];
      if (c + 1 < chunks) __builtin_prefetch(gsrc + 16 * PLD, 0, 0);
    }
    __syncthreads();
    const float* myLds = &buf[0][0];
#endif

    // A fragments for gaussian m = lane&15: unconditional vector loads
    // (b128 + b64, both naturally aligned; row stride 48B) then register
    // selects -- avoids exec-masked scalar ds_loads in the hot loop.
    const float* Fm = myLds + n * PLD;
    v4f f03 = *(const v4f*)(Fm);      // F0..F3
    v2f f45 = *(const v2f*)(Fm + 4);  // F4..F5
    v2f a1, a2;
    a1.x = hi ? f03.z : f03.x;
    a1.y = hi ? f03.w : f03.y;
    a2.x = hi ? 0.0f : f45.x;
    a2.y = hi ? 0.0f : f45.y;

    v8f cc = {};
    cc = __builtin_amdgcn_wmma_f32_16x16x4_f32(false, a1, false, b1,
                                               (short)0, cc, false, false);
    cc = __builtin_amdgcn_wmma_f32_16x16x4_f32(false, a2, false, b2,
                                               (short)0, cc, false, false);

    // chunk-local front-to-back scan over this lane-half's 8 gaussians
    float localT = 1.0f, lr = 0, lg = 0, lb = 0, ldp = 0, lw = 0;
#pragma unroll
    for (int r = 0; r < 8; ++r) {
      const float* row = myLds + (rbase + r) * PLD;
      float pw = cc[r];
      float al = fminf(0.99f, row[9] * __expf(pw));
      bool ok = (pw <= 0.0f) && (al >= (1.0f / 255.0f));
      al = ok ? al : 0.0f;
      float wgt = al * localT;
      localT *= (1.0f - al);
      lr += wgt * row[6];
      lg += wgt * row[7];
      lb += wgt * row[8];
      ldp += wgt * row[10];
      lw += wgt;
    }
    // merge lane halves: hi half comes after lo half in depth order
    float otherProd = __shfl_xor(localT, 16, 32);
    float factor = hi ? (T * otherProd) : T;
    accR += factor * lr;
    accG += factor * lg;
    accB += factor * lb;
    accD += factor * ldp;
    accW += factor * lw;
    T *= localT * otherProd;

#if HAS_TDM
    if (wave == 0) __builtin_amdgcn_s_wait_tensorcnt(0);
    __syncthreads();
#else
    __syncthreads();
#endif
  }

  // fold partner-lane partial sums (pixel shared by lane l and l+16)
  accR += __shfl_xor(accR, 16, 32);
  accG += __shfl_xor(accG, 16, 32);
  accB += __shfl_xor(accB, 16, 32);
  accD += __shfl_xor(accD, 16, 32);
  accW += __shfl_xor(accW, 16, 32);

  if (!hi && active) {
    float bgr = bg[b * 3 + 0], bgg = bg[b * 3 + 1], bgb = bg[b * 3 + 2];
    size_t bp = (size_t)b * 3;
    out_img[(bp + 0) * P + p] = accR + T * bgr;
    out_img[(bp + 1) * P + p] = accG + T * bgg;
    out_img[(bp + 2) * P + p] = accB + T * bgb;
    out_depth[(size_t)b * P + p] = accD;
    out_op[(size_t)b * P + p] = accW;
  }
}

// ---------------------------------------------------------------------------
extern "C" void kernel_launch(void* const* d_in, const int* in_sizes, int n_in,
                              void* d_out, int out_size, void* d_ws,
                              size_t ws_size, hipStream_t stream) {
  const float* extr  = (const float*)d_in[0];
  const float* intr  = (const float*)d_in[1];
  const float* nearv = (const float*)d_in[2];
  // d_in[3] = far (unused by reference output)
  const int* hPtr = (const int*)d_in[4];
  const int* wPtr = (const int*)d_in[5];
  const float* bg    = (const float*)d_in[6];
  const float* means = (const float*)d_in[7];
  const float* covs  = (const float*)d_in[8];
  const float* shs   = (const float*)d_in[9];
  const float* opac  = (const float*)d_in[10];

  int B = in_sizes[2];             // near has B elements
  int G = in_sizes[10] / B;        // opacities: B*G
  int P = out_size / (5 * B);      // outputs: B*(3+1+1)*H*W
  int Npow2 = 16;
  while (Npow2 < G) Npow2 <<= 1;   // bitonic size (assumes G <= 1024)

  float* payload = (float*)d_ws;                         // B*G*PLD
  float* keys    = payload + (size_t)B * G * PLD;        // B*G
  float* sorted  = keys + (size_t)B * G;                 // B*Npow2*PLD

  float* out       = (float*)d_out;
  float* out_img   = out;
  float* out_depth = out + (size_t)B * 3 * P;
  float* out_op    = out_depth + (size_t)B * P;

  int total = B * G;
  preprocess_kernel<<<(total + 255) / 256, 256, 0, stream>>>(
      extr, intr, nearv, hPtr, wPtr, means, covs, shs, opac, payload, keys, B, G);

  sort_kernel<<<B, Npow2, 0, stream>>>(keys, payload, sorted, G);

  int blocksPerBatch = (P + 127) / 128;  // 8 waves x 16 pixels per block
  render_kernel<<<B * blocksPerBatch, 256, 0, stream>>>(
      sorted, bg, wPtr, out_img, out_depth, out_op, Npow2, P, blocksPerBatch);
}